// Attention_67319317397635
// MI455X (gfx1250) — compile-verified
//
#include <hip/hip_runtime.h>
#include <hip/hip_bf16.h>

// ---------------------------------------------------------------------------
// MI455X (gfx1250) fused linear-attention kernel.
//
// Pipeline: K1 (WMMA f32 GEMM + elementwise q/k + per-block kv partials)
//        -> K2 (kv partial reduction, deterministic, no atomics)
//        -> K3 (x = q*kv, sigmoid, * input1)
// ---------------------------------------------------------------------------

typedef __attribute__((ext_vector_type(2))) float v2f;
typedef __attribute__((ext_vector_type(8))) float v8f;

namespace {
constexpr int kB   = 4;
constexpr int kN   = 16384;
constexpr int kD   = 256;            // DIM1 == DIM2
constexpr int kH   = 8;
constexpr int kHD  = 32;
constexpr long kRows = (long)kB * kN;        // 65536
constexpr int kRowsPerBlock = 128;           // 8 waves * 16 rows
constexpr int kGemmBlocks = (int)(kRows / kRowsPerBlock);  // 512
constexpr int kKvPerBH = kHD * kHD;          // 1024
constexpr int kKvPerB  = kH * kKvPerBH;      // 8192
}

// ---------------------------------------------------------------------------
// K1: per wave: 16-row band.
//   GEMM: q_in[16x256] = input1[16x256] @ conv_w^T via v_wmma_f32_16x16x4_f32
//   Then elementwise on fragments (each 16-lane half-wave owns one row at a
//   time: lane m holds cols {16t+m}), q written out, k stashed back into the
//   accumulator registers, then per-head 32x32 kv partials in registers,
//   reduced across the 8 waves through LDS (no atomics => deterministic).
// ---------------------------------------------------------------------------
__global__ __launch_bounds__(256, 1)
void fused_gemm_qk_kv(const float* __restrict__ input1,
                      const float* __restrict__ input2,
                      const float* __restrict__ conv_w,
                      const float* __restrict__ pos_enc,
                      const float* __restrict__ scale_p,
                      float* __restrict__ q_out,
                      float* __restrict__ kv_part)
{
    __shared__ float s_isc[kD];              // 1/softplus(scale_p)
    __shared__ float s_red[8 * kKvPerBH];    // 32 KB cross-wave reduce buffer

    const int tid  = threadIdx.x;
    const int wave = tid >> 5;
    const int lane = tid & 31;
    const int m    = lane & 15;              // position within 16-lane half
    const int half = lane >> 4;              // which half of the wave

    if (tid < kD) {
        float sp = scale_p[tid];
        s_isc[tid] = 1.0f / log1pf(__expf(sp));   // 1/softplus
    }
    __syncthreads();

    const long rbase = (long)blockIdx.x * kRowsPerBlock + (long)wave * 16;

    // ---------------- GEMM: 16 col-tiles of 16, K = 256 in steps of 4 ------
    v8f acc[16];
    #pragma unroll
    for (int t = 0; t < 16; ++t) {
        #pragma unroll
        for (int i = 0; i < 8; ++i) acc[t][i] = 0.0f;
    }

    // A frag: lane holds A[m][k0 + 2*half + {0,1}]  (ISA 16x4 f32 layout)
    // B frag: lane holds B[k0 + 2*half + {0,1}][n=16t+m], B[k][n]=conv_w[n*256+k]
    const float* arow = input1 + (rbase + m) * kD + half * 2;
    const float* bcol = conv_w + (long)m * kD + half * 2;
    for (int k0 = 0; k0 < kD; k0 += 4) {
        v2f a = *(const v2f*)(arow + k0);
        #pragma unroll
        for (int t = 0; t < 16; ++t) {
            v2f bf = *(const v2f*)(bcol + (long)t * 16 * kD + k0);
            acc[t] = __builtin_amdgcn_wmma_f32_16x16x4_f32(
                false, a, false, bf, (short)0, acc[t], false, false);
        }
    }

    // ---------------- elementwise q/k on fragments --------------------------
    // C layout: VGPR r: lanes 0-15 -> row r, lanes 16-31 -> row r+8; col 16t+m.
    #pragma unroll
    for (int r = 0; r < 8; ++r) {
        const long gr = rbase + r + 8 * half;       // this half-wave's row
        const int  n  = (int)(gr & (kN - 1));
        const float* posr = pos_enc + (long)n * kD;
        const float* in2r = input2 + gr * kD;

        float q3[16], k3[16];
        float sq1 = 0.f, sq2 = 0.f, sk1 = 0.f, sk2 = 0.f;
        #pragma unroll
        for (int t = 0; t < 16; ++t) {
            const int c = t * 16 + m;
            float p   = posr[c];
            float isc = s_isc[c];
            float q = fmaxf(acc[t][r] + p, 0.0f) * isc;
            float k = fmaxf(in2r[c]  + p, 0.0f) * isc;
            sq1 += q * q;  sk1 += k * k;
            float qc = q * q * q, kc = k * k * k;
            sq2 += qc * qc;  sk2 += kc * kc;
            q3[t] = qc;  k3[t] = kc;
        }
        // reduce the 4 sums across the 16-lane half (xor < 16 stays in-half)
        #pragma unroll
        for (int s = 1; s < 16; s <<= 1) {
            sq1 += __shfl_xor(sq1, s, 32);
            sq2 += __shfl_xor(sq2, s, 32);
            sk1 += __shfl_xor(sk1, s, 32);
            sk2 += __shfl_xor(sk2, s, 32);
        }
        const float qfac = sqrtf(sq1) * rsqrtf(sq2);  // ||q|| / ||q^3||
        const float kfac = sqrtf(sk1) * rsqrtf(sk2);

        float* qo = q_out + gr * kD;
        #pragma unroll
        for (int t = 0; t < 16; ++t) {
            qo[t * 16 + m] = qfac * q3[t];
            acc[t][r]      = kfac * k3[t];    // stash k_final in freed accums
        }
    }

    // ---------------- kv partials: per head, 32x32 over this wave's rows ----
    // k(h, d) for a row lives in acc[2h + d/16][r] of lane (16*half + d%16).
    // Lane owns e = m and e = 16+m columns of the outer product.
    #pragma unroll
    for (int h = 0; h < kH; ++h) {
        float a0[32], a1[32];
        #pragma unroll
        for (int d = 0; d < 32; ++d) { a0[d] = 0.f; a1[d] = 0.f; }

        #pragma unroll
        for (int r = 0; r < 8; ++r) {
            const long gr = rbase + r + 8 * half;
            const float* vr = input2 + gr * kD + h * kHD;   // v = raw input2
            float v0 = vr[m];
            float v1 = vr[16 + m];
            #pragma unroll
            for (int d = 0; d < 32; ++d) {
                float kd = __shfl(acc[2 * h + (d >> 4)][r],
                                  16 * half + (d & 15), 32);
                a0[d] += kd * v0;
                a1[d] += kd * v1;
            }
        }
        // fold the two half-wave partials together (rows r and r+8)
        #pragma unroll
        for (int d = 0; d < 32; ++d) {
            a0[d] += __shfl_xor(a0[d], 16, 32);
            a1[d] += __shfl_xor(a1[d], 16, 32);
        }
        if (half == 0) {
            #pragma unroll
            for (int d = 0; d < 32; ++d) {
                s_red[wave * kKvPerBH + d * 32 + m]      = a0[d];
                s_red[wave * kKvPerBH + d * 32 + 16 + m] = a1[d];
            }
        }
        __syncthreads();
        // reduce the 8 wave partials -> per-block partial for this head
        for (int i = tid; i < kKvPerBH; i += 256) {
            float s = 0.f;
            #pragma unroll
            for (int w = 0; w < 8; ++w) s += s_red[w * kKvPerBH + i];
            kv_part[(long)blockIdx.x * kKvPerB + h * kKvPerBH + i] = s;
        }
        __syncthreads();
    }
}

// ---------------------------------------------------------------------------
// K2: kv[b,h,d,e] = (1/N) * sum over the 128 blocks of batch b.
// ---------------------------------------------------------------------------
__global__ __launch_bounds__(256, 1)
void kv_reduce(const float* __restrict__ kv_part, float* __restrict__ kv)
{
    const int idx = blockIdx.x * 256 + threadIdx.x;   // 0..32767, grid=128
    const int b = idx >> 13;                          // / kKvPerB
    const int i = idx & (kKvPerB - 1);
    const float* p = kv_part + (long)b * 128 * kKvPerB + i;
    float s = 0.f;
    for (int blk = 0; blk < 128; ++blk) s += p[(long)blk * kKvPerB];
    kv[idx] = s * (1.0f / (float)kN);
}

// ---------------------------------------------------------------------------
// K3: x[row, h*32+e] = sum_d q[row, h*32+d] * kv[b,h,d,e];
//     out = sigmoid(x) * input1.   kv[b] staged in LDS (32 KB).
// ---------------------------------------------------------------------------
__global__ __launch_bounds__(256, 1)
void apply_out(const float* __restrict__ q,
               const float* __restrict__ kv,
               const float* __restrict__ input1,
               float* __restrict__ out)
{
    __shared__ float s_kv[kKvPerB];
    constexpr int kRowsPB = 64;                       // 8 waves * 8 rows
    const long rbase = (long)blockIdx.x * kRowsPB;    // grid = 1024
    const int  b = (int)(rbase >> 14);                // / kN

    for (int i = threadIdx.x; i < kKvPerB; i += 256)
        s_kv[i] = kv[(long)b * kKvPerB + i];
    __syncthreads();

    const int wave = threadIdx.x >> 5;
    const int lane = threadIdx.x & 31;

    for (int rr = 0; rr < 8; ++rr) {
        const long row = rbase + (long)wave * 8 + rr;
        const float* qr = q + row * kD;
        const float* i1 = input1 + row * kD;
        float* o = out + row * kD;

        float qreg[8];                                // q(head j, d=lane)
        #pragma unroll
        for (int j = 0; j < 8; ++j) qreg[j] = qr[j * 32 + lane];

        #pragma unroll
        for (int j = 0; j < 8; ++j) {
            float x = 0.f;
            #pragma unroll
            for (int d = 0; d < 32; ++d) {
                float qd = __shfl(qreg[j], d, 32);
                x += qd * s_kv[j * kKvPerBH + d * 32 + lane];
            }
            float sig = 1.0f / (1.0f + __expf(-x));
            o[j * 32 + lane] = sig * i1[j * 32 + lane];
        }
    }
}

// ---------------------------------------------------------------------------
extern "C" void kernel_launch(void* const* d_in, const int* in_sizes, int n_in,
                              void* d_out, int out_size, void* d_ws, size_t ws_size,
                              hipStream_t stream)
{
    const float* input1  = (const float*)d_in[0];   // [4,16384,256]
    const float* input2  = (const float*)d_in[1];   // [4,16384,256]
    const float* conv_w  = (const float*)d_in[2];   // [256,256]
    const float* pos_enc = (const float*)d_in[3];   // [1,16384,256]
    const float* scale_p = (const float*)d_in[4];   // [1,1,256]
    float* out = (float*)d_out;

    float* ws_q    = (float*)d_ws;                       // 64 MB
    float* ws_part = ws_q + kRows * kD;                  // 16 MB
    float* ws_kv   = ws_part + (long)kGemmBlocks * kKvPerB;  // 128 KB

    fused_gemm_qk_kv<<<kGemmBlocks, 256, 0, stream>>>(
        input1, input2, conv_w, pos_enc, scale_p, ws_q, ws_part);
    kv_reduce<<<(kB * kKvPerB) / 256, 256, 0, stream>>>(ws_part, ws_kv);
    apply_out<<<(int)(kRows / 64), 256, 0, stream>>>(ws_q, ws_kv, input1, out);
}